// net_55800215109702
// MI455X (gfx1250) — compile-verified
//
#include <hip/hip_runtime.h>
#include <hip/hip_bf16.h>

typedef float v2f __attribute__((ext_vector_type(2)));
typedef float v8f __attribute__((ext_vector_type(8)));

#define IN_F 73
#define EPS 1e-5f
#define SLOPE 0.1f

// ---------------------------------------------------------------- utilities
__global__ void zero_kernel(float* __restrict__ p, long long n) {
    long long i = (long long)blockIdx.x * blockDim.x + threadIdx.x;
    long long stride = (long long)gridDim.x * blockDim.x;
    for (; i < n; i += stride) p[i] = 0.0f;
}

// ------------------------------------------------------- SpMM (one TAG hop)
// h_out[dst] += w * h_in[src], edge-parallel: one wave per edge, lanes cover
// the 73 features. Hardware fp32 global atomics; the 29MB feature table is
// L2-resident on MI455X's 192MB L2, so gather/scatter mostly hits L2.
__global__ void __launch_bounds__(256)
spmm_kernel(const float* __restrict__ hin, const int* __restrict__ src,
            const int* __restrict__ dst, const float* __restrict__ ew,
            float* __restrict__ hout, int E) {
    int warp  = (blockIdx.x * blockDim.x + threadIdx.x) >> 5;
    int lane  = threadIdx.x & 31;
    int nwarp = (gridDim.x * blockDim.x) >> 5;
    for (int e = warp; e < E; e += nwarp) {
        // stream-ahead prefetch of the edge lists (only non-L2-resident data)
        int ep = e + 4 * nwarp;
        if (ep < E) {
            __builtin_prefetch(&src[ep], 0, 1);
            __builtin_prefetch(&dst[ep], 0, 1);
            __builtin_prefetch(&ew[ep], 0, 1);
        }
        int s = src[e];
        int d = dst[e];
        float w = ew[e];
        const float* hs = hin + (long long)s * IN_F;
        float* hd = hout + (long long)d * IN_F;
        for (int f = lane; f < IN_F; f += 32) {
            unsafeAtomicAdd(&hd[f], hs[f] * w);
        }
    }
}

// --------------------------------------------- concat(x,h1,h2,h3) @ W_tag
// fp32 WMMA GEMM: D(16x16) += A(16x4) x B(4x16), K = 292 = 4 arrays * 73.
// Per source array: 18 divergence-free K=4 steps (cols 0..71, uniform base
// pointer). The 4 leftover columns (col 72 of each array) are packed into one
// final K=4 step: lanes<16 -> K={0,1} from {x,h1}, lanes>=16 -> K={2,3} from
// {h2,h3} (pointer cndmask only, EXEC stays all-1s as WMMA requires).
// A frag (32-bit 16x4): VGPR0 = K even, VGPR1 = K odd; lane half selects K pair.
// D frag (32-bit 16x16): VGPR r: lanes 0-15 -> M=r, lanes 16-31 -> M=8+r.
__global__ void __launch_bounds__(128)
tag_gemm_wmma(const float* __restrict__ x,  const float* __restrict__ h1,
              const float* __restrict__ h2, const float* __restrict__ h3,
              const float* __restrict__ Wt, const float* __restrict__ bt,
              float* __restrict__ htag, int N) {
    const int wave = threadIdx.x >> 5;
    const int lane = threadIdx.x & 31;
    const int rowTile = blockIdx.x * 2 + (wave >> 1);
    const int colTile = wave & 1;
    const bool hi = lane >= 16;
    const int l15 = lane & 15;

    int mrow = rowTile * 16 + l15;
    if (mrow >= N) mrow = N - 1;           // clamp: EXEC must stay all-1s
    const int ncol = colTile * 16 + l15;   // 0..31
    const long long rbase = (long long)mrow * IN_F;

    const float* pr0 = x  + rbase;
    const float* pr1 = h1 + rbase;
    const float* pr2 = h2 + rbase;
    const float* pr3 = h3 + rbase;

    v8f acc = {};
    const int kh = hi ? 2 : 0;

#pragma unroll
    for (int arr = 0; arr < 4; ++arr) {
        const float* p = (arr == 0) ? pr0 : (arr == 1) ? pr1 : (arr == 2) ? pr2 : pr3;
        const float* wp = Wt + arr * IN_F * 32 + ncol;
#pragma unroll
        for (int k0 = 0; k0 < 72; k0 += 4) {
            const int ka = k0 + kh;
            v2f a, b;
            a.x = p[ka];
            a.y = p[ka + 1];
            b.x = wp[ka * 32];
            b.y = wp[(ka + 1) * 32];
            acc = __builtin_amdgcn_wmma_f32_16x16x4_f32(
                false, a, false, b, (short)0, acc, false, false);
        }
    }

    // tail: K rows = col 72 of {x, h1, h2, h3} <-> B rows {72,145,218,291}
    {
        const float* pa = hi ? pr2 : pr0;   // K = 0 / 2
        const float* pb = hi ? pr3 : pr1;   // K = 1 / 3
        v2f a, b;
        a.x = pa[72];
        a.y = pb[72];
        b.x = Wt[(hi ? (3 * IN_F - 1) : (IN_F - 1)) * 32 + ncol];        // 218 / 72
        b.y = Wt[(hi ? (4 * IN_F - 1) : (2 * IN_F - 1)) * 32 + ncol];    // 291 / 145
        acc = __builtin_amdgcn_wmma_f32_16x16x4_f32(
            false, a, false, b, (short)0, acc, false, false);
    }

    const float bb = bt[ncol];
#pragma unroll
    for (int r = 0; r < 8; ++r) {
        int m = rowTile * 16 + (hi ? 8 : 0) + r;
        if (m < N) htag[(long long)m * 32 + ncol] = acc[r] + bb;
    }
}

// -------------------------------------------------------- batch statistics
// st[0..F) = sum, st[F..2F) = sum of squares (block-reduced, then global atomics)
template <int F>
__global__ void __launch_bounds__(256)
stats_kernel(const float* __restrict__ z, int N, float* __restrict__ st) {
    __shared__ float ssum[F];
    __shared__ float ssq[F];
    const int t = threadIdx.x;
    if (t < F) { ssum[t] = 0.0f; ssq[t] = 0.0f; }
    __syncthreads();
    const int f = t % F;
    const int rpb = 256 / F;
    const int stride = gridDim.x * rpb;
    float s = 0.0f, q = 0.0f;
    for (int r = blockIdx.x * rpb + t / F; r < N; r += stride) {
        float v = z[(long long)r * F + f];
        s += v;
        q += v * v;
    }
    atomicAdd(&ssum[f], s);
    atomicAdd(&ssq[f], q);
    __syncthreads();
    if (t < F) {
        atomicAdd(&st[t], ssum[t]);
        atomicAdd(&st[F + t], ssq[t]);
    }
}

// ---------------------------------------- fused BN(train) + LeakyReLU + Linear
template <int FIN, int FOUT>
__global__ void __launch_bounds__(256)
bn_lin_kernel(const float* __restrict__ zin, int N,
              const float* __restrict__ st, const float* __restrict__ g,
              const float* __restrict__ be, const float* __restrict__ W,
              const float* __restrict__ bias, float* __restrict__ zout) {
    int r = blockIdx.x * blockDim.x + threadIdx.x;
    if (r >= N) return;
    const float invN = 1.0f / (float)N;
    float a[FIN];
#pragma unroll
    for (int i = 0; i < FIN; ++i) {
        float m = st[i] * invN;
        float v = st[FIN + i] * invN - m * m;
        float t = (zin[(long long)r * FIN + i] - m) * rsqrtf(v + EPS) * g[i] + be[i];
        a[i] = (t >= 0.0f) ? t : SLOPE * t;
    }
#pragma unroll
    for (int j = 0; j < FOUT; ++j) {
        float acc = bias[j];
#pragma unroll
        for (int i = 0; i < FIN; ++i) acc += a[i] * W[i * FOUT + j];
        zout[(long long)r * FOUT + j] = acc;
    }
}

// ---------------------------------------------- final BN + LReLU + softmax(2)
__global__ void __launch_bounds__(256)
bn_softmax_kernel(const float* __restrict__ z, int N,
                  const float* __restrict__ st, const float* __restrict__ g,
                  const float* __restrict__ be, float* __restrict__ out) {
    int r = blockIdx.x * blockDim.x + threadIdx.x;
    if (r >= N) return;
    const float invN = 1.0f / (float)N;
    float a[2];
#pragma unroll
    for (int i = 0; i < 2; ++i) {
        float m = st[i] * invN;
        float v = st[2 + i] * invN - m * m;
        float t = (z[(long long)r * 2 + i] - m) * rsqrtf(v + EPS) * g[i] + be[i];
        a[i] = (t >= 0.0f) ? t : SLOPE * t;
    }
    float mx = fmaxf(a[0], a[1]);
    float e0 = __expf(a[0] - mx);
    float e1 = __expf(a[1] - mx);
    float inv = 1.0f / (e0 + e1);
    out[(long long)r * 2 + 0] = e0 * inv;
    out[(long long)r * 2 + 1] = e1 * inv;
}

// ---------------------------------------------------------------- launcher
extern "C" void kernel_launch(void* const* d_in, const int* in_sizes, int n_in,
                              void* d_out, int out_size, void* d_ws, size_t ws_size,
                              hipStream_t stream) {
    const float* x    = (const float*)d_in[0];
    const int*   esrc = (const int*)d_in[1];
    const int*   edst = (const int*)d_in[2];
    const float* ew   = (const float*)d_in[3];
    const float* Wt   = (const float*)d_in[4];
    const float* bt   = (const float*)d_in[5];
    const float* W1   = (const float*)d_in[6];
    const float* b1   = (const float*)d_in[7];
    const float* W2   = (const float*)d_in[8];
    const float* b2   = (const float*)d_in[9];
    const float* W3   = (const float*)d_in[10];
    const float* b3   = (const float*)d_in[11];
    const float* W4   = (const float*)d_in[12];
    const float* b4   = (const float*)d_in[13];
    const float* g1   = (const float*)d_in[14];
    const float* be1  = (const float*)d_in[15];
    const float* g2   = (const float*)d_in[16];
    const float* be2  = (const float*)d_in[17];
    const float* g3   = (const float*)d_in[18];
    const float* be3  = (const float*)d_in[19];
    const float* g4   = (const float*)d_in[20];
    const float* be4  = (const float*)d_in[21];
    const float* g5   = (const float*)d_in[22];
    const float* be5  = (const float*)d_in[23];

    const int N = in_sizes[0] / IN_F;
    const int E = in_sizes[3];

    // workspace layout (floats)
    float* ws = (float*)d_ws;
    long long off = 0;
    float* h1   = ws + off; off += (long long)N * IN_F;
    float* h2   = ws + off; off += (long long)N * IN_F;
    float* h3   = ws + off; off += (long long)N * IN_F;
    float* htag = ws + off; off += (long long)N * 32;
    float* z1   = ws + off; off += (long long)N * 16;
    float* z2   = ws + off; off += (long long)N * 8;
    float* z3   = ws + off; off += (long long)N * 4;
    float* z4   = ws + off; off += (long long)N * 2;
    float* st1  = ws + off; off += 64;
    float* st2  = ws + off; off += 32;
    float* st3  = ws + off; off += 16;
    float* st4  = ws + off; off += 8;
    float* st5  = ws + off; off += 4;

    // zero scatter targets + stats accumulators (graph-capture safe)
    zero_kernel<<<2048, 256, 0, stream>>>(h1, 3LL * N * IN_F);
    zero_kernel<<<1, 128, 0, stream>>>(st1, 64 + 32 + 16 + 8 + 4);

    // K=3 propagation hops: h_k = A_w @ h_{k-1}
    spmm_kernel<<<2048, 256, 0, stream>>>(x,  esrc, edst, ew, h1, E);
    spmm_kernel<<<2048, 256, 0, stream>>>(h1, esrc, edst, ew, h2, E);
    spmm_kernel<<<2048, 256, 0, stream>>>(h2, esrc, edst, ew, h3, E);

    // [x|h1|h2|h3] @ W_tag + b_tag  via v_wmma_f32_16x16x4_f32 (73 WMMAs/wave)
    tag_gemm_wmma<<<(N + 31) / 32, 128, 0, stream>>>(x, h1, h2, h3, Wt, bt, htag, N);

    const int nb = (N + 255) / 256;
    stats_kernel<32><<<256, 256, 0, stream>>>(htag, N, st1);
    bn_lin_kernel<32, 16><<<nb, 256, 0, stream>>>(htag, N, st1, g1, be1, W1, b1, z1);
    stats_kernel<16><<<256, 256, 0, stream>>>(z1, N, st2);
    bn_lin_kernel<16, 8><<<nb, 256, 0, stream>>>(z1, N, st2, g2, be2, W2, b2, z2);
    stats_kernel<8><<<256, 256, 0, stream>>>(z2, N, st3);
    bn_lin_kernel<8, 4><<<nb, 256, 0, stream>>>(z2, N, st3, g3, be3, W3, b3, z3);
    stats_kernel<4><<<256, 256, 0, stream>>>(z3, N, st4);
    bn_lin_kernel<4, 2><<<nb, 256, 0, stream>>>(z3, N, st4, g4, be4, W4, b4, z4);
    stats_kernel<2><<<256, 256, 0, stream>>>(z4, N, st5);
    bn_softmax_kernel<<<nb, 256, 0, stream>>>(z4, N, st5, g5, be5, (float*)d_out);
}